// PoolFeatureMapping_60687887892523
// MI455X (gfx1250) — compile-verified
//
#include <hip/hip_runtime.h>

// ---------------------------------------------------------------------------
// Sparse 2x2x2 max-pool (segment max) + gather to b-sites.  0 FLOPs, ~300 MB
// traffic -> WMMA inapplicable; this is a memory-system kernel.
// MI455X strategy:
//   * 64 MB pooled scratch is kept resident in the 192 MB L2 (RT stores,
//     device-scope atomics resolve at L2, gather re-reads from L2).
//   * Float max via MONOTONE UNSIGNED KEY encoding -> one non-divergent
//     global_atomic_max_u32 per lane (vs. divergent i32-max/u32-min pair),
//     and the empty-segment sentinel becomes 0.
//   * wave32-aware layout: 64 lanes per row => row index is wave-uniform;
//     readfirstlane scalarizes coordinate loads to s_load.
//   * NT hints on the read-once feature stream and write-once output; b128
//     vector accesses; grid-stride loops with global_prefetch_b8.
// ---------------------------------------------------------------------------

typedef float    v4f __attribute__((ext_vector_type(4)));
typedef unsigned v4u __attribute__((ext_vector_type(4)));

#define CCH 64          // channels per row (fixed by reference)
#define TPB 256         // 8 waves per block (wave32)

// Order-preserving map from IEEE-754 float bits to unsigned:
//   b >= 0 : b | 0x80000000   (top bit set)
//   b <  0 : ~b               (top bit clear, reversed order)
// => float max == unsigned max over keys; 0 is below every finite key.
__device__ __forceinline__ unsigned encode_key(float v) {
    unsigned b = __float_as_uint(v);
    unsigned m = (unsigned)((int)b >> 31);          // all-ones if v < 0
    return b ^ (m | 0x80000000u);
}

__device__ __forceinline__ float decode_key(unsigned k) {
    unsigned m = (unsigned)((int)k >> 31);          // all-ones if key from v >= 0
    unsigned bits = k ^ (~m | 0x80000000u);
    float f = __uint_as_float(bits);
    return (k != 0u) ? f : 0.0f;                    // 0 key == empty segment
}

// Phase 1: zero-fill pooled key scratch [Dp^3 * 64] (0 = "empty" sentinel).
// Regular-temporal b128 stores keep the 64 MB buffer dirty-resident in L2.
__global__ void pool_init_kernel(unsigned* __restrict__ pooled,
                                 const int* __restrict__ spatial) {
    const int D  = *spatial;                        // uniform scalar load
    const int Dp = D >> 1;
    const long total4 = (long)Dp * Dp * Dp * (CCH / 4);
    const v4u zero = { 0u, 0u, 0u, 0u };
    v4u* p = (v4u*)pooled;
    const long stride = (long)gridDim.x * blockDim.x;
    for (long i = (long)blockIdx.x * blockDim.x + threadIdx.x; i < total4; i += stride) {
        p[i] = zero;                                // global_store_b128 (stay in L2)
    }
}

// Phase 2: scatter-max.  One lane per (row, channel); 64 lanes per row, so
// row is wave-uniform -> readfirstlane turns coord loads into s_load.
// Feature stream: NT b32 load.  Max: single global_atomic_max_u32 (L2-res).
__global__ void pool_scatter_kernel(const float* __restrict__ feats,
                                    const int* __restrict__ acoords,
                                    unsigned* __restrict__ pooled,
                                    const int* __restrict__ spatial,
                                    long total /* = NA*64 */) {
    const int D  = *spatial;
    const int Dp = D >> 1;
    const long stride = (long)gridDim.x * blockDim.x;
    for (long g = (long)blockIdx.x * blockDim.x + threadIdx.x; g < total; g += stride) {
        // prefetch next grid-stride iteration of the feature stream
        // (global_prefetch_b8; speculative, OOB silently dropped)
        __builtin_prefetch(feats + g + stride, 0, 3);
        // g>>6 is wave-uniform (wave spans 32 consecutive g inside one
        // 64-aligned block) -> force scalarization of the coord loads.
        const int row = __builtin_amdgcn_readfirstlane((int)(g >> 6));
        const int c   = (int)(g & 63);
        const int x = acoords[row * 3 + 0] >> 1;    // s_load (uniform addr)
        const int y = acoords[row * 3 + 1] >> 1;
        const int z = acoords[row * 3 + 2] >> 1;
        const int seg = (x * Dp + y) * Dp + z;
        const float v = __builtin_nontemporal_load(feats + g);   // read-once stream
        atomicMax(pooled + (long)seg * CCH + c, encode_key(v));  // one u32-max atomic
    }
}

// Phase 3: gather.  16 lanes x uint4 per b-row: contiguous 256 B L2 reads of
// the pooled key row, branchless decode (+ empty->0), NT b128 output stores.
__global__ void pool_gather_kernel(const int* __restrict__ bcoords,
                                   const unsigned* __restrict__ pooled,
                                   float* __restrict__ out,
                                   const int* __restrict__ spatial,
                                   long total /* = NB*16 */) {
    const int D  = *spatial;
    const int Dp = D >> 1;
    const v4u* pooled4 = (const v4u*)pooled;
    v4f* out4 = (v4f*)out;
    const long stride = (long)gridDim.x * blockDim.x;
    for (long g = (long)blockIdx.x * blockDim.x + threadIdx.x; g < total; g += stride) {
        // prefetch next iteration's coord word
        __builtin_prefetch(bcoords + ((g + stride) >> 4) * 3, 0, 3);
        const int row = (int)(g >> 4);
        const int q   = (int)(g & 15);
        const int x = bcoords[row * 3 + 0] >> 1;
        const int y = bcoords[row * 3 + 1] >> 1;
        const int z = bcoords[row * 3 + 2] >> 1;
        const int seg = (x * Dp + y) * Dp + z;
        const v4u k = pooled4[(long)seg * (CCH / 4) + q];   // L2-resident b128 read
        v4f r;
        r.x = decode_key(k.x);
        r.y = decode_key(k.y);
        r.z = decode_key(k.z);
        r.w = decode_key(k.w);
        __builtin_nontemporal_store(r, out4 + (long)row * (CCH / 4) + q);
    }
}

extern "C" void kernel_launch(void* const* d_in, const int* in_sizes, int n_in,
                              void* d_out, int out_size, void* d_ws, size_t ws_size,
                              hipStream_t stream) {
    const float* a_feats  = (const float*)d_in[0];
    const int*   a_coords = (const int*)d_in[1];
    const int*   b_coords = (const int*)d_in[2];
    const int*   spatial  = (const int*)d_in[3];   // device scalar (D)
    float*    out    = (float*)d_out;
    unsigned* pooled = (unsigned*)d_ws;            // (D/2)^3 * 64 keys (64 MB @ D=128)

    const long NA = in_sizes[1] / 3;               // a_coords is [NA,3]
    const long NB = in_sizes[2] / 3;               // b_coords is [NB,3]

    // Phase 1: zero the pooled key scratch (grid-stride; D only known on device)
    pool_init_kernel<<<4096, TPB, 0, stream>>>(pooled, spatial);

    // Phase 2: scatter-max, ~4 grid-stride iterations per thread
    {
        const long total = NA * CCH;
        int blocks = (int)((total + (long)TPB * 4 - 1) / ((long)TPB * 4));
        if (blocks < 1) blocks = 1;
        pool_scatter_kernel<<<blocks, TPB, 0, stream>>>(a_feats, a_coords, pooled,
                                                        spatial, total);
    }

    // Phase 3: gather to output, uint4/float4 granularity
    {
        const long total = NB * (CCH / 4);
        int blocks = (int)((total + (long)TPB * 4 - 1) / ((long)TPB * 4));
        if (blocks < 1) blocks = 1;
        pool_gather_kernel<<<blocks, TPB, 0, stream>>>(b_coords, pooled, out,
                                                       spatial, total);
    }
}